// MFGNN_88098369176052
// MI455X (gfx1250) — compile-verified
//
#include <hip/hip_runtime.h>

typedef float v2f __attribute__((ext_vector_type(2)));
typedef float v8f __attribute__((ext_vector_type(8)));

// ---------------------------------------------------------------------------
// copy src[N,F] (dense) into dst[N,ldo] starting at dst column offset baked
// into the dst pointer.  One thread per element.
// ---------------------------------------------------------------------------
__global__ void copy_strided_kernel(const float* __restrict__ src, int F,
                                    float* __restrict__ dst, int ldo,
                                    long long total) {
    long long i = (long long)blockIdx.x * blockDim.x + threadIdx.x;
    if (i >= total) return;
    long long row = i / F;
    int col = (int)(i - row * F);
    dst[row * (long long)ldo + col] = src[i];
}

__global__ void zero_kernel(float* __restrict__ p, long long n) {
    long long i = (long long)blockIdx.x * blockDim.x + threadIdx.x;
    if (i < n) p[i] = 0.0f;
}

// ---------------------------------------------------------------------------
// Segment-sum: agg[dst[e], :] += feat[src[e], :].
// One thread per (edge, float4-chunk).  Consecutive threads cover consecutive
// chunks of the same edge -> coalesced gather; scatter is L2 f32 atomics.
// All node tensors + edge list fit in the 192MB L2, so this is L2-resident.
// ---------------------------------------------------------------------------
__global__ void scatter_add_kernel(const float* __restrict__ feat,
                                   const long long* __restrict__ src,
                                   const long long* __restrict__ dst,
                                   float* __restrict__ agg,
                                   long long E, int Fc /* = F/4 */, int F) {
    long long t = (long long)blockIdx.x * blockDim.x + threadIdx.x;
    if (t >= E * (long long)Fc) return;
    long long e = t / Fc;
    int c = (int)(t - e * Fc);
    long long s = src[e];
    long long d = dst[e];
    const float4 v = *(const float4*)(feat + s * (long long)F + c * 4);
    float* p = agg + d * (long long)F + c * 4;
    atomicAdd(p + 0, v.x);
    atomicAdd(p + 1, v.y);
    atomicAdd(p + 2, v.z);
    atomicAdd(p + 3, v.w);
}

// ---------------------------------------------------------------------------
// Fused GraphConv GEMM using native fp32 WMMA (V_WMMA_F32_16X16X4_F32):
//   out[16x16 tile] = act( A_tile @ Wrel + X_tile @ Wroot + bias (+ extra) )
// One wave per (16-row, 16-col) output tile. Accumulates both GEMMs into the
// same v8f accumulator.  FIN/FOUT compile-time -> fully unrolled wmma chain.
//
// Fragment layouts (wave32, ISA 7.12.2):
//   A 16x4 f32 : lane<16 -> row=lane,  a={K0,K1}; lane>=16 -> row=lane-16, a={K2,K3}
//   B 4x16 f32 : lane<16 -> col=lane,  b={K0,K1}; lane>=16 -> col=lane-16, b={K2,K3}
//   C/D 16x16  : vgpr r   -> M = r + 8*(lane/16), N = lane%16
// ---------------------------------------------------------------------------
template <int FIN, int FOUT, int RELU, int HAS_EXTRA>
__global__ void gconv_gemm_kernel(const float* __restrict__ A,      // [N, FIN] aggregated
                                  const float* __restrict__ X,      // [N, FIN] root
                                  const float* __restrict__ Wrel,   // [FIN, FOUT]
                                  const float* __restrict__ Wroot,  // [FIN, FOUT]
                                  const float* __restrict__ bias,   // [FOUT]
                                  const float* __restrict__ extra,  // [N, FOUT] or null
                                  float* __restrict__ out,          // [N, ldo]
                                  int N, int ldo) {
    constexpr int TILES_N = FOUT / 16;
    const int wave = (blockIdx.x * blockDim.x + threadIdx.x) >> 5;
    const int lane = threadIdx.x & 31;
    const int tiles_m = N >> 4;                 // N assumed multiple of 16
    const int m_tile = wave / TILES_N;
    const int n_tile = wave - m_tile * TILES_N;
    if (m_tile >= tiles_m) return;              // wave-uniform: EXEC all-ones inside

    const int row0 = m_tile << 4;
    const int n0 = n_tile << 4;
    const int half = lane >> 4;                 // 0: K lo pair, 1: K hi pair
    const int l16 = lane & 15;
    const long long arow = row0 + l16;          // A/X row for this lane
    const int ncol = n0 + l16;                  // B/out column for this lane

    v8f acc = {};

    // rel path:  A @ Wrel
#pragma unroll
    for (int k0 = 0; k0 < FIN; k0 += 4) {
        const int ka = k0 + (half << 1);
        v2f a = *(const v2f*)(A + arow * FIN + ka);          // 8B aligned (ka even)
        v2f b;
        b.x = Wrel[(long long)ka * FOUT + ncol];
        b.y = Wrel[(long long)(ka + 1) * FOUT + ncol];
        acc = __builtin_amdgcn_wmma_f32_16x16x4_f32(false, a, false, b,
                                                    (short)0, acc, false, false);
    }
    // root path: X @ Wroot  (summed into the same accumulator)
#pragma unroll
    for (int k0 = 0; k0 < FIN; k0 += 4) {
        const int ka = k0 + (half << 1);
        v2f a = *(const v2f*)(X + arow * FIN + ka);
        v2f b;
        b.x = Wroot[(long long)ka * FOUT + ncol];
        b.y = Wroot[(long long)(ka + 1) * FOUT + ncol];
        acc = __builtin_amdgcn_wmma_f32_16x16x4_f32(false, a, false, b,
                                                    (short)0, acc, false, false);
    }

    const float bn = bias[ncol];
#pragma unroll
    for (int r = 0; r < 8; ++r) {
        const long long m = row0 + r + (half << 3);
        float v = acc[r] + bn;
        if (HAS_EXTRA) v += extra[m * FOUT + ncol];
        if (RELU) v = fmaxf(v, 0.0f);
        out[m * (long long)ldo + ncol] = v;
    }
}

// ---------------------------------------------------------------------------

static inline long long cdivll(long long a, long long b) { return (a + b - 1) / b; }

extern "C" void kernel_launch(void* const* d_in, const int* in_sizes, int n_in,
                              void* d_out, int out_size, void* d_ws, size_t ws_size,
                              hipStream_t stream) {
    const float*     x   = (const float*)d_in[0];      // [N,48]
    const long long* ei  = (const long long*)d_in[1];  // [2,E] int64
    const float*     ax  = (const float*)d_in[2];      // [N,16]
    const float*     lf  = (const float*)d_in[3];      // [N,16]
    const float*     W1r = (const float*)d_in[4];      // [64,32]
    const float*     b1  = (const float*)d_in[5];      // [32]
    const float*     W1o = (const float*)d_in[6];      // [64,32]
    const float*     W2r = (const float*)d_in[7];      // [48,48]
    const float*     b2  = (const float*)d_in[8];      // [48]
    const float*     W2o = (const float*)d_in[9];      // [48,48]
    const float*     W3r = (const float*)d_in[10];     // [48,16]
    const float*     b3  = (const float*)d_in[11];     // [16]
    const float*     W3o = (const float*)d_in[12];     // [48,16]
    float*           out = (float*)d_out;              // [N,16]

    const long long N = in_sizes[0] / 48;
    const long long E = in_sizes[1] / 2;
    const long long* src = ei;
    const long long* dst = ei + E;

    // workspace layout (floats): h0[N,64] | agg[N,64] | h1c[N,48] | h2[N,48]
    float* h0  = (float*)d_ws;
    float* agg = h0 + N * 64;
    float* h1c = agg + N * 64;
    float* h2  = h1c + N * 48;

    const int T = 256;

    // ---- h0 = concat(x, additional_x) -> [N,64]
    copy_strided_kernel<<<(int)cdivll(N * 48, T), T, 0, stream>>>(x, 48, h0, 64, N * 48);
    copy_strided_kernel<<<(int)cdivll(N * 16, T), T, 0, stream>>>(ax, 16, h0 + 48, 64, N * 16);

    // ---- layer 1: 64 -> 32, relu; write into cols [0,32) of h1c (ldo=48)
    zero_kernel<<<(int)cdivll(N * 64, T), T, 0, stream>>>(agg, N * 64);
    scatter_add_kernel<<<(int)cdivll(E * 16, T), T, 0, stream>>>(h0, src, dst, agg, E, 16, 64);
    {
        long long waves = (N / 16) * (32 / 16);
        gconv_gemm_kernel<64, 32, 1, 0><<<(int)cdivll(waves * 32, T), T, 0, stream>>>(
            agg, h0, W1r, W1o, b1, nullptr, h1c, (int)N, 48);
    }
    // ---- h1c cols [32,48) = lf_embedding
    copy_strided_kernel<<<(int)cdivll(N * 16, T), T, 0, stream>>>(lf, 16, h1c + 32, 48, N * 16);

    // ---- layer 2: 48 -> 48, relu
    zero_kernel<<<(int)cdivll(N * 48, T), T, 0, stream>>>(agg, N * 48);
    scatter_add_kernel<<<(int)cdivll(E * 12, T), T, 0, stream>>>(h1c, src, dst, agg, E, 12, 48);
    {
        long long waves = (N / 16) * (48 / 16);
        gconv_gemm_kernel<48, 48, 1, 0><<<(int)cdivll(waves * 32, T), T, 0, stream>>>(
            agg, h1c, W2r, W2o, b2, nullptr, h2, (int)N, 48);
    }

    // ---- layer 3: 48 -> 16, no relu, + additional_x, -> d_out
    zero_kernel<<<(int)cdivll(N * 48, T), T, 0, stream>>>(agg, N * 48);
    scatter_add_kernel<<<(int)cdivll(E * 12, T), T, 0, stream>>>(h2, src, dst, agg, E, 12, 48);
    {
        long long waves = (N / 16) * (16 / 16);
        gconv_gemm_kernel<48, 16, 0, 1><<<(int)cdivll(waves * 32, T), T, 0, stream>>>(
            agg, h2, W3r, W3o, b3, ax, out, (int)N, 16);
    }
}